// MaskedGCNLayer_61529701483171
// MI455X (gfx1250) — compile-verified
//
#include <hip/hip_runtime.h>

// MaskedGCNLayer for MI455X (gfx1250, wave32).
// B=8, N=1024, E=16384, Din=U=128. f32 throughout: reference is f32 and the
// problem is L2/HBM-bandwidth bound (2.1 GFLOP total), so the exact-precision
// V_WMMA_F32_16X16X4_F32 matrix path is the right choice.
//
// Factorization: out = mask .* ( diag(inv) * ( A @ (diag(inv) * (x@W)) ) + bias )
//  - column scaling diag(inv) fused into the support-GEMM epilogue
//  - row scaling diag(inv) fused into the bias/mask epilogue
//  => main GEMM inner loop is a pure A-load/B-load/WMMA stream.

#define B_   8
#define N_   1024
#define E_   16384
#define DIN_ 128
#define U_   128
#define EPS_ 1e-6f

typedef __attribute__((ext_vector_type(2))) float v2f;
typedef __attribute__((ext_vector_type(8))) float v8f;

// ---------------------------------------------------------------- zero ws ---
__global__ void k_zero4(float4* __restrict__ p, long n4) {
    long i = (long)blockIdx.x * blockDim.x + threadIdx.x;
    if (i < n4) p[i] = make_float4(0.f, 0.f, 0.f, 0.f);
}

// ------------------------------------------------------- edge scatter-add ---
__global__ void k_scatter(const int* __restrict__ edge_index,
                          const unsigned char* __restrict__ edge_mask,
                          float* __restrict__ adj) {
    int idx = blockIdx.x * blockDim.x + threadIdx.x;
    if (idx >= B_ * E_) return;
    int b = idx >> 14;            // E_ = 16384
    int e = idx & (E_ - 1);
    const int* ei = edge_index + (size_t)b * 2 * E_;
    int src = ei[e];
    src = src < 0 ? 0 : (src > N_ - 1 ? N_ - 1 : src);
    int dst = ei[E_ + e];
    dst = dst < 0 ? 0 : (dst > N_ - 1 ? N_ - 1 : dst);
    if (edge_mask[(size_t)b * E_ + e])
        atomicAdd(&adj[((size_t)b * N_ + dst) * N_ + src], 1.0f);
}

// ------------------------------------------------------------- self loops ---
__global__ void k_selfloop(const unsigned char* __restrict__ node_mask,
                           float* __restrict__ adj) {
    int idx = blockIdx.x * blockDim.x + threadIdx.x;   // idx = b*N + n
    if (idx >= B_ * N_) return;
    int n = idx & (N_ - 1);
    adj[(size_t)idx * N_ + n] += (float)node_mask[idx];
}

// --------------------------------------- degree row-sum + rsqrt (1 wave/row)
__global__ void k_degree(const float* __restrict__ adj,
                         float* __restrict__ inv) {
    int row  = blockIdx.x * 8 + (threadIdx.x >> 5);    // 8 waves per block
    int lane = threadIdx.x & 31;
    const float* arow = adj + (size_t)row * N_;
    float s = 0.f;
    for (int j = lane; j < N_; j += 32) s += arow[j];
    #pragma unroll
    for (int o = 16; o > 0; o >>= 1) s += __shfl_down(s, o, 32);
    if (lane == 0) inv[row] = rsqrtf(fmaxf(s, EPS_));
}

// ------------- support = diag(inv) * (x @ kernel)   (8192 x 128 x 128) ------
// One wave per 16x16 tile; 32 V_WMMA_F32_16X16X4_F32 per tile; the per-row
// inv_sqrt_deg column-scaling of the main GEMM is fused into the epilogue.
__global__ void k_gemm_support(const float* __restrict__ x,
                               const float* __restrict__ kern,
                               const float* __restrict__ inv,
                               float* __restrict__ support) {
    int gwave = (blockIdx.x * blockDim.x + threadIdx.x) >> 5;  // 0..4095
    int lane  = threadIdx.x & 31;
    int tn = gwave & 7;           // 8 N-tiles
    int tm = gwave >> 3;          // 512 M-tiles
    int m0 = tm << 4, n0 = tn << 4;
    int half = lane >> 4;         // 0: K={0,1}, 1: K={2,3}
    int l16  = lane & 15;

    const float* Arow = x + (size_t)(m0 + l16) * DIN_ + 2 * half; // row-major A
    const float* Bcol = kern + n0 + l16;                          // kern[k][u]

    v8f c = {0.f, 0.f, 0.f, 0.f, 0.f, 0.f, 0.f, 0.f};
    #pragma unroll 4
    for (int k = 0; k < DIN_; k += 4) {
        int kk = k + 2 * half;
        v2f a = *(const v2f*)(Arow + k);
        v2f bb; bb.x = Bcol[(size_t)kk * U_]; bb.y = Bcol[(size_t)(kk + 1) * U_];
        c = __builtin_amdgcn_wmma_f32_16x16x4_f32(
                false, a, false, bb, (short)0, c, false, false);
    }
    int col = n0 + l16;
    #pragma unroll
    for (int r = 0; r < 8; ++r) {
        int row = m0 + r + half * 8;      // C layout: VGPR r -> M=r / M=r+8
        support[(size_t)row * U_ + col] = inv[row] * c[r];
    }
}

// ----------------- acc = A @ support_scaled, per batch (1024x1024x128) ------
// One wave per 64x16 output block: 4 accumulators share each B fragment
// (4x reuse of support traffic; A slice is L2-resident: 32MB vs 192MB L2).
// Inner loop: 4x global_load_b64 (A) + 2x global_load_b32 (B) + 4x WMMA.
__global__ void k_gemm_main(const float* __restrict__ adj,
                            const float* __restrict__ sups,
                            float* __restrict__ out) {
    int gwave = (blockIdx.x * blockDim.x + threadIdx.x) >> 5;  // 0..1023
    int lane  = threadIdx.x & 31;
    int b  = gwave >> 7;          // 128 waves per batch
    int t  = gwave & 127;
    int tn = t & 7;               // 8 N-tiles (U=128)
    int tmb = t >> 3;             // 16 M-blocks of 64 rows
    int m0 = tmb << 6, n0 = tn << 4;
    int half = lane >> 4;
    int l16  = lane & 15;

    const float* A = adj  + (size_t)b * N_ * N_;
    const float* S = sups + (size_t)b * N_ * U_;

    const float* A0   = A + (size_t)(m0 + l16) * N_ + 2 * half;  // rows +0/16/32/48
    const float* Scol = S + n0 + l16;

    v8f c0 = {0.f,0.f,0.f,0.f,0.f,0.f,0.f,0.f};
    v8f c1 = c0, c2 = c0, c3 = c0;

    for (int k = 0; k < N_; k += 4) {
        int kk = k + 2 * half;
        __builtin_prefetch(A0 + k + 256, 0, 0);       // global_prefetch on A stream
        v2f bb;
        bb.x = Scol[(size_t)kk * U_];
        bb.y = Scol[(size_t)(kk + 1) * U_];
        v2f a0 = *(const v2f*)(A0 + k);
        v2f a1 = *(const v2f*)(A0 + k + 16 * N_);     // 64KB imm offsets
        v2f a2 = *(const v2f*)(A0 + k + 32 * N_);
        v2f a3 = *(const v2f*)(A0 + k + 48 * N_);
        c0 = __builtin_amdgcn_wmma_f32_16x16x4_f32(false, a0, false, bb, (short)0, c0, false, false);
        c1 = __builtin_amdgcn_wmma_f32_16x16x4_f32(false, a1, false, bb, (short)0, c1, false, false);
        c2 = __builtin_amdgcn_wmma_f32_16x16x4_f32(false, a2, false, bb, (short)0, c2, false, false);
        c3 = __builtin_amdgcn_wmma_f32_16x16x4_f32(false, a3, false, bb, (short)0, c3, false, false);
    }

    int col = n0 + l16;
    float* ob = out + (size_t)b * N_ * U_;
    #pragma unroll
    for (int r = 0; r < 8; ++r) {
        int row = m0 + r + half * 8;
        ob[(size_t)(row     ) * U_ + col] = c0[r];
        ob[(size_t)(row + 16) * U_ + col] = c1[r];
        ob[(size_t)(row + 32) * U_ + col] = c2[r];
        ob[(size_t)(row + 48) * U_ + col] = c3[r];
    }
}

// ----------------------- epilogue: mask * (inv_row * acc + bias) ------------
__global__ void k_finish(const unsigned char* __restrict__ node_mask,
                         const float* __restrict__ bias,
                         const float* __restrict__ inv,
                         float* __restrict__ out) {
    int i = blockIdx.x * blockDim.x + threadIdx.x;
    if (i >= B_ * N_ * U_) return;
    int u  = i & (U_ - 1);
    int bn = i >> 7;
    float m = (float)node_mask[bn];
    out[i] = m * (inv[bn] * out[i] + bias[u]);
}

extern "C" void kernel_launch(void* const* d_in, const int* in_sizes, int n_in,
                              void* d_out, int out_size, void* d_ws, size_t ws_size,
                              hipStream_t stream) {
    const float*         x          = (const float*)d_in[0];
    const unsigned char* node_mask  = (const unsigned char*)d_in[1];
    const int*           edge_index = (const int*)d_in[2];
    const unsigned char* edge_mask  = (const unsigned char*)d_in[3];
    const float*         kern       = (const float*)d_in[4];
    const float*         bias       = (const float*)d_in[5];
    float*               out        = (float*)d_out;

    // workspace layout (floats): adj[B*N*N] | support[B*N*U] | inv[B*N]
    float* adj     = (float*)d_ws;
    float* support = adj + (size_t)B_ * N_ * N_;
    float* inv     = support + (size_t)B_ * N_ * U_;

    long n4 = (long)B_ * N_ * N_ / 4;
    k_zero4<<<(int)((n4 + 255) / 256), 256, 0, stream>>>((float4*)adj, n4);

    k_scatter<<<(B_ * E_ + 255) / 256, 256, 0, stream>>>(edge_index, edge_mask, adj);
    k_selfloop<<<(B_ * N_ + 255) / 256, 256, 0, stream>>>(node_mask, adj);
    k_degree<<<(B_ * N_) / 8, 256, 0, stream>>>(adj, inv);

    // support GEMM: 4096 tiles, 8 waves (256 thr) per block -> 512 blocks
    k_gemm_support<<<512, 256, 0, stream>>>(x, kern, inv, support);
    // main GEMM: 1024 waves (64x16 blocks) -> 128 blocks
    k_gemm_main<<<128, 256, 0, stream>>>(adj, support, out);

    k_finish<<<(B_ * N_ * U_ + 255) / 256, 256, 0, stream>>>(node_mask, bias, inv, out);
}